// HGTEncoder_1236950581297
// MI455X (gfx1250) — compile-verified
//
#include <hip/hip_runtime.h>
#include <math.h>

// ---------------------------------------------------------------------------
// HGT forward for MI455X (gfx1250, wave32, WMMA).
// - All dense algebra via V_WMMA_F32_16X16X4_F32 (full fp32; reference is fp32).
// - Edge relation transforms hoisted from edges to source nodes.
// - Segment softmax in TWO edge sweeps: (A) logits + atomic max,
//   (BC) exp + atomic den + unnormalized V scatter; then a fused
//   node-level normalize+GELU pass. Saves a full edge sweep vs 3-pass.
// ---------------------------------------------------------------------------

typedef __attribute__((ext_vector_type(2))) float v2f;
typedef __attribute__((ext_vector_type(8))) float v8f;

// ---------------- WMMA GEMM: C[M,N] = A[M,K] @ B[K,N] + bias[N] -------------
// One wave (32 lanes) computes one 16x16 tile of C, stepping K by 4.
// A fragment (ISA 7.12.2, 32-bit A 16x4): m = lane&15; lane<16 -> K={kk,kk+1},
// lane>=16 -> K={kk+2,kk+3}. B fragment (4x16): n = lane&15, VGPR r ->
// K = kk + r + 2*(lane>>4). C tile: VGPR r -> row r + 8*(lane>>4), col lane&15.
__global__ void wmma_gemm_bias(const float* __restrict__ A, int lda,
                               const float* __restrict__ B, int ldb,
                               const float* __restrict__ bias,
                               float* __restrict__ C, int ldc,
                               int M, int N, int K) {
  const int wavesPerBlock = blockDim.x >> 5;
  const int waveId = blockIdx.x * wavesPerBlock + (threadIdx.x >> 5);
  const int lane = threadIdx.x & 31;
  const int tilesM = (M + 15) >> 4;
  const int tilesN = N >> 4;            // N is always a multiple of 16 here
  if (waveId >= tilesM * tilesN) return;
  const int tm = waveId / tilesN;
  const int tn = waveId % tilesN;

  const int half = lane >> 4;           // 0 or 1
  const int l16  = lane & 15;
  const int koff = half << 1;           // 0 or 2
  int rowA = (tm << 4) + l16;
  if (rowA > M - 1) rowA = M - 1;       // clamp (reads only; stores guarded)
  const int colB = (tn << 4) + l16;

  v8f c = {};
  for (int kk = 0; kk < K; kk += 4) {
    v2f a, b;
    a.x = A[(size_t)rowA * lda + kk + koff];
    a.y = A[(size_t)rowA * lda + kk + koff + 1];
    b.x = B[(size_t)(kk + koff) * ldb + colB];
    b.y = B[(size_t)(kk + koff + 1) * ldb + colB];
    c = __builtin_amdgcn_wmma_f32_16x16x4_f32(false, a, false, b,
                                              (short)0, c, false, false);
  }
  const float bv = bias ? bias[colB] : 0.0f;
#pragma unroll
  for (int r = 0; r < 8; ++r) {
    const int row = (tm << 4) + r + (half << 3);
    if (row < M) C[(size_t)row * ldc + colB] = c[r] + bv;
  }
}

// -------- Per-head relation transform: OUT[n, h*32+f] = sum_d KQV[n, off+h*32+d] * R[h,d,f]
// grid.y = head. N = 32 -> 2 column tiles per head. K = 32 -> 8 WMMA steps.
__global__ void wmma_rel_gemm(const float* __restrict__ KQV, int kqvStride, int srcColOff,
                              const float* __restrict__ R,   // [H][32][32]
                              float* __restrict__ OUT,       // [M][128]
                              int M) {
  const int h = blockIdx.y;
  const int wavesPerBlock = blockDim.x >> 5;
  const int waveId = blockIdx.x * wavesPerBlock + (threadIdx.x >> 5);
  const int lane = threadIdx.x & 31;
  const int tilesM = (M + 15) >> 4;
  if (waveId >= tilesM * 2) return;
  const int tm = waveId >> 1;
  const int tn = waveId & 1;

  const int half = lane >> 4;
  const int l16  = lane & 15;
  const int koff = half << 1;
  int rowA = (tm << 4) + l16;
  if (rowA > M - 1) rowA = M - 1;
  const int colB = (tn << 4) + l16;

  const float* Ah = KQV + srcColOff + h * 32;
  const float* Bh = R + h * 32 * 32;

  v8f c = {};
#pragma unroll
  for (int kk = 0; kk < 32; kk += 4) {
    v2f a, b;
    a.x = Ah[(size_t)rowA * kqvStride + kk + koff];
    a.y = Ah[(size_t)rowA * kqvStride + kk + koff + 1];
    b.x = Bh[(kk + koff) * 32 + colB];
    b.y = Bh[(kk + koff + 1) * 32 + colB];
    c = __builtin_amdgcn_wmma_f32_16x16x4_f32(false, a, false, b,
                                              (short)0, c, false, false);
  }
#pragma unroll
  for (int r = 0; r < 8; ++r) {
    const int row = (tm << 4) + r + (half << 3);
    if (row < M) OUT[(size_t)row * 128 + h * 32 + colB] = c[r];
  }
}

// ---------------- float atomic max via monotone integer encoding ------------
__device__ inline void atomicMaxFloat(float* addr, float value) {
  if (value >= 0.0f)
    atomicMax((int*)addr, __float_as_int(value));
  else
    atomicMin((unsigned int*)addr, __float_as_uint(value));
}

// ---------------- Pass A: attention logits + segment max --------------------
// One wave per edge: lanes cooperatively read the 512B kt row and 512B q row
// (fully coalesced float4), dot per head (8 lanes/head), shfl-reduce.
__global__ void edge_attn(const int* __restrict__ src, const int* __restrict__ dst,
                          const float* __restrict__ kt,     // [Nsrc][128]
                          const float* __restrict__ kqvDst, // [Ndst][384], q at +128
                          const float* __restrict__ prel,   // [H]
                          float* __restrict__ alpha,        // [E][4]
                          float* __restrict__ amax,         // [Ndst][4]
                          int E) {
  const int wavesPerBlock = blockDim.x >> 5;
  const int e = blockIdx.x * wavesPerBlock + (threadIdx.x >> 5);
  if (e >= E) return;
  const int lane = threadIdx.x & 31;
  const int s = src[e], d = dst[e];
  const float4 kv = *(const float4*)(kt + (size_t)s * 128 + lane * 4);
  const float4 qv = *(const float4*)(kqvDst + (size_t)d * 384 + 128 + lane * 4);
  float p = kv.x * qv.x + kv.y * qv.y + kv.z * qv.z + kv.w * qv.w;
  p += __shfl_xor(p, 1, 32);
  p += __shfl_xor(p, 2, 32);
  p += __shfl_xor(p, 4, 32);
  const int h = lane >> 3;
  if ((lane & 7) == 0) {
    const float a = p * prel[h] * 0.17677669529663687f;   // 1/sqrt(32)
    alpha[(size_t)e * 4 + h] = a;
    atomicMaxFloat(&amax[(size_t)d * 4 + h], a);
  }
}

// ------- Pass BC: exp, atomic den accumulate, UNNORMALIZED V scatter --------
// One wave per edge. Normalization (division by den) is deferred to the
// node-level norm_gelu pass, saving a full edge sweep.
__global__ void edge_softmax_scatter(const int* __restrict__ src, const int* __restrict__ dst,
                                     const float* __restrict__ vt,    // [Nsrc][128]
                                     const float* __restrict__ alpha, // [E][4] raw logits
                                     const float* __restrict__ amax,  // [Ndst][4]
                                     float* __restrict__ den,         // [Ndst][4]
                                     float* __restrict__ agg,         // [Ndst][128]
                                     int E) {
  const int wavesPerBlock = blockDim.x >> 5;
  const int e = blockIdx.x * wavesPerBlock + (threadIdx.x >> 5);
  if (e >= E) return;
  const int lane = threadIdx.x & 31;
  const int s = src[e], d = dst[e];
  const int h = lane >> 3;
  const float ex = __expf(alpha[(size_t)e * 4 + h] - amax[(size_t)d * 4 + h]);
  if ((lane & 7) == 0) unsafeAtomicAdd(&den[(size_t)d * 4 + h], ex);
  const float4 v = *(const float4*)(vt + (size_t)s * 128 + lane * 4);
  float* out = agg + (size_t)d * 128 + lane * 4;
  unsafeAtomicAdd(out + 0, ex * v.x);
  unsafeAtomicAdd(out + 1, ex * v.y);
  unsafeAtomicAdd(out + 2, ex * v.z);
  unsafeAtomicAdd(out + 3, ex * v.w);
}

// -------- Node-level: agg /= den (softmax denominator), then GELU -----------
__global__ void norm_gelu(float* __restrict__ agg, const float* __restrict__ den,
                          long long n /* nodes*128 */) {
  long long i = (long long)blockIdx.x * blockDim.x + threadIdx.x;
  if (i < n) {
    const long long node = i >> 7;
    const int h = (int)(i & 127) >> 5;
    const float x = agg[i] / den[node * 4 + h];
    const float t = tanhf(0.7978845608028654f * (x + 0.044715f * x * x * x));
    agg[i] = 0.5f * x * (1.0f + t);
  }
}

// ---------------- elementwise helpers ---------------------------------------
__global__ void fill_f32(float* __restrict__ p, float v, long long n) {
  long long i = (long long)blockIdx.x * blockDim.x + threadIdx.x;
  if (i < n) p[i] = v;
}

__global__ void mix_skip(const float* __restrict__ o, const float* __restrict__ xin,
                         const float* __restrict__ skip, int l,
                         float* __restrict__ xout, long long n) {
  long long i = (long long)blockIdx.x * blockDim.x + threadIdx.x;
  if (i < n) {
    const float g = 1.0f / (1.0f + __expf(-skip[l]));
    xout[i] = g * o[i] + (1.0f - g) * xin[i];
  }
}

// ---------------------------------------------------------------------------
extern "C" void kernel_launch(void* const* d_in, const int* in_sizes, int n_in,
                              void* d_out, int out_size, void* d_ws, size_t ws_size,
                              hipStream_t stream) {
  (void)n_in; (void)out_size; (void)ws_size;

  const float* x_d    = (const float*)d_in[0];
  const float* x_g    = (const float*)d_in[1];
  const int*  src_dg  = (const int*)d_in[2];
  const int*  dst_dg  = (const int*)d_in[3];
  const int*  src_gd  = (const int*)d_in[4];
  const int*  dst_gd  = (const int*)d_in[5];
  const int*  src_gg  = (const int*)d_in[6];
  const int*  dst_gg  = (const int*)d_in[7];
  const float* Wkqv_d = (const float*)d_in[8];
  const float* bkqv_d = (const float*)d_in[9];
  const float* Wkqv_g = (const float*)d_in[10];
  const float* bkqv_g = (const float*)d_in[11];
  const float* Wout_d = (const float*)d_in[12];
  const float* bout_d = (const float*)d_in[13];
  const float* Wout_g = (const float*)d_in[14];
  const float* bout_g = (const float*)d_in[15];
  const float* skip_d = (const float*)d_in[16];
  const float* skip_g = (const float*)d_in[17];
  const float* arel   = (const float*)d_in[18];
  const float* mrel   = (const float*)d_in[19];
  const float* prel   = (const float*)d_in[20];

  const int HD = 128, H = 4;
  const int ND = in_sizes[0] / HD;
  const int NG = in_sizes[1] / HD;
  const int E_dg = in_sizes[2];
  const int E_gd = in_sizes[4];
  const int E_gg = in_sizes[6];
  const int L = in_sizes[16];

  // ----- workspace layout (floats). Zero-init buffers are contiguous, as are
  //       the -inf-init amax buffers, so each layer needs only two fills. -----
  float* ws = (float*)d_ws;
  auto alloc = [&](size_t n) { float* p = ws; ws += n; return p; };
  float* kqv_d  = alloc((size_t)ND * 384);
  float* kqv_g  = alloc((size_t)NG * 384);
  float* kt_dg  = alloc((size_t)ND * 128);
  float* vt_dg  = alloc((size_t)ND * 128);
  float* kt_gd  = alloc((size_t)NG * 128);
  float* vt_gd  = alloc((size_t)NG * 128);
  float* kt_gg  = alloc((size_t)NG * 128);
  float* vt_gg  = alloc((size_t)NG * 128);
  float* alpha  = alloc(((size_t)E_dg + E_gd + E_gg) * H);
  float* amax_d = alloc((size_t)ND * H);           // contiguous -inf region
  float* amax_g = alloc((size_t)NG * H);
  float* den_d  = alloc((size_t)ND * H);           // contiguous zero region
  float* den_g  = alloc((size_t)NG * H);
  float* agg_d  = alloc((size_t)ND * 128);
  float* agg_g  = alloc((size_t)NG * 128);
  float* o_d    = alloc((size_t)ND * 128);
  float* o_g    = alloc((size_t)NG * 128);
  float* xbuf_d[2] = { alloc((size_t)ND * 128), alloc((size_t)ND * 128) };
  float* xbuf_g[2] = { alloc((size_t)NG * 128), alloc((size_t)NG * 128) };

  const long long nInf  = (long long)(ND + NG) * H;                 // amax_d..amax_g
  const long long nZero = (long long)(ND + NG) * H
                        + (long long)(ND + NG) * 128;               // den_d..agg_g

  float* out_d = (float*)d_out;
  float* out_g = (float*)d_out + (size_t)ND * HD;

  float* alpha_dg = alpha;
  float* alpha_gd = alpha + (size_t)E_dg * H;
  float* alpha_gg = alpha + ((size_t)E_dg + E_gd) * H;

  const dim3 blk(256);
  auto gemmBlocks = [](int M, int N) {
    long long waves = (long long)((M + 15) / 16) * (N / 16);
    return dim3((unsigned)((waves + 7) / 8));
  };
  auto relBlocks = [](int M) {
    long long waves = (long long)((M + 15) / 16) * 2;
    return dim3((unsigned)((waves + 7) / 8), 4);
  };
  auto edgeBlocks = [](int E) { return dim3((unsigned)((E + 7) / 8)); };
  auto eltBlocks  = [](long long n) { return dim3((unsigned)((n + 255) / 256)); };

  const float* xd_in = x_d;
  const float* xg_in = x_g;

  for (int l = 0; l < L; ++l) {
    float* xd_out = (l == L - 1) ? out_d : xbuf_d[l & 1];
    float* xg_out = (l == L - 1) ? out_g : xbuf_g[l & 1];

    // 1) fused K/Q/V projection (WMMA f32)
    wmma_gemm_bias<<<gemmBlocks(ND, 384), blk, 0, stream>>>(
        xd_in, 128, Wkqv_d + (size_t)l * 128 * 384, 384,
        bkqv_d + (size_t)l * 384, kqv_d, 384, ND, 384, 128);
    wmma_gemm_bias<<<gemmBlocks(NG, 384), blk, 0, stream>>>(
        xg_in, 128, Wkqv_g + (size_t)l * 128 * 384, 384,
        bkqv_g + (size_t)l * 384, kqv_g, 384, NG, 384, 128);

    // 2) per-head relation transforms at source nodes (WMMA f32)
    const size_t relStride = (size_t)H * 32 * 32;       // per (l,et)
    const float* A0 = arel + ((size_t)l * 3 + 0) * relStride;
    const float* A1 = arel + ((size_t)l * 3 + 1) * relStride;
    const float* A2 = arel + ((size_t)l * 3 + 2) * relStride;
    const float* M0 = mrel + ((size_t)l * 3 + 0) * relStride;
    const float* M1 = mrel + ((size_t)l * 3 + 1) * relStride;
    const float* M2 = mrel + ((size_t)l * 3 + 2) * relStride;
    wmma_rel_gemm<<<relBlocks(ND), blk, 0, stream>>>(kqv_d, 384,   0, A0, kt_dg, ND);
    wmma_rel_gemm<<<relBlocks(ND), blk, 0, stream>>>(kqv_d, 384, 256, M0, vt_dg, ND);
    wmma_rel_gemm<<<relBlocks(NG), blk, 0, stream>>>(kqv_g, 384,   0, A1, kt_gd, NG);
    wmma_rel_gemm<<<relBlocks(NG), blk, 0, stream>>>(kqv_g, 384, 256, M1, vt_gd, NG);
    wmma_rel_gemm<<<relBlocks(NG), blk, 0, stream>>>(kqv_g, 384,   0, A2, kt_gg, NG);
    wmma_rel_gemm<<<relBlocks(NG), blk, 0, stream>>>(kqv_g, 384, 256, M2, vt_gg, NG);

    // 3) reset softmax stats + aggregates (two contiguous fills)
    fill_f32<<<eltBlocks(nInf),  blk, 0, stream>>>(amax_d, -INFINITY, nInf);
    fill_f32<<<eltBlocks(nZero), blk, 0, stream>>>(den_d, 0.0f, nZero);

    // 4) pass A: logits + segment max (across ALL edge types per dst)
    const float* p_dg = prel + ((size_t)l * 3 + 0) * H;
    const float* p_gd = prel + ((size_t)l * 3 + 1) * H;
    const float* p_gg = prel + ((size_t)l * 3 + 2) * H;
    edge_attn<<<edgeBlocks(E_dg), blk, 0, stream>>>(src_dg, dst_dg, kt_dg, kqv_g, p_dg, alpha_dg, amax_g, E_dg);
    edge_attn<<<edgeBlocks(E_gd), blk, 0, stream>>>(src_gd, dst_gd, kt_gd, kqv_d, p_gd, alpha_gd, amax_d, E_gd);
    edge_attn<<<edgeBlocks(E_gg), blk, 0, stream>>>(src_gg, dst_gg, kt_gg, kqv_g, p_gg, alpha_gg, amax_g, E_gg);

    // 5) pass BC: exp + den accumulate + unnormalized V scatter
    edge_softmax_scatter<<<edgeBlocks(E_dg), blk, 0, stream>>>(src_dg, dst_dg, vt_dg, alpha_dg, amax_g, den_g, agg_g, E_dg);
    edge_softmax_scatter<<<edgeBlocks(E_gd), blk, 0, stream>>>(src_gd, dst_gd, vt_gd, alpha_gd, amax_d, den_d, agg_d, E_gd);
    edge_softmax_scatter<<<edgeBlocks(E_gg), blk, 0, stream>>>(src_gg, dst_gg, vt_gg, alpha_gg, amax_g, den_g, agg_g, E_gg);

    // 6) node-level normalize (divide by den) + GELU, in place
    norm_gelu<<<eltBlocks((long long)ND * 128), blk, 0, stream>>>(agg_d, den_d, (long long)ND * 128);
    norm_gelu<<<eltBlocks((long long)NG * 128), blk, 0, stream>>>(agg_g, den_g, (long long)NG * 128);

    // 7) output projection (WMMA f32)
    wmma_gemm_bias<<<gemmBlocks(ND, 128), blk, 0, stream>>>(
        agg_d, 128, Wout_d + (size_t)l * 128 * 128, 128,
        bout_d + (size_t)l * 128, o_d, 128, ND, 128, 128);
    wmma_gemm_bias<<<gemmBlocks(NG, 128), blk, 0, stream>>>(
        agg_g, 128, Wout_g + (size_t)l * 128 * 128, 128,
        bout_g + (size_t)l * 128, o_g, 128, NG, 128, 128);

    // 8) sigmoid-gated skip mix
    mix_skip<<<eltBlocks((long long)ND * 128), blk, 0, stream>>>(o_d, xd_in, skip_d, l, xd_out, (long long)ND * 128);
    mix_skip<<<eltBlocks((long long)NG * 128), blk, 0, stream>>>(o_g, xg_in, skip_g, l, xg_out, (long long)NG * 128);

    xd_in = xd_out;
    xg_in = xg_out;
  }
}